// HGTLayer_41798621724771
// MI455X (gfx1250) — compile-verified
//
#include <hip/hip_runtime.h>
#include <math.h>

#define NN 50000
#define EE 800000
#define INF_ 256
#define OUTF 256
#define HH 8
#define DKK 32
#define LDS_STRIDE 260   // 16-row X tile, padded so 16 rows hit distinct banks

typedef __attribute__((ext_vector_type(2))) float v2f;
typedef __attribute__((ext_vector_type(8))) float v8f;

// ---------------------------------------------------------------- utilities
__global__ void fill_kernel(float* __restrict__ p, float v, int n) {
    int i = blockIdx.x * blockDim.x + threadIdx.x;
    if (i < n) p[i] = v;
}

__device__ inline void atomicMaxF(float* addr, float val) {
    int old = __float_as_int(*addr);
    while (val > __int_as_float(old)) {
        int assumed = old;
        old = atomicCAS((int*)addr, assumed, __float_as_int(val));
        if (old == assumed) break;
    }
}

// ------------------------------------------------- fold rel transforms into W
// Wf[m][i][h*32+c] = sum_d W[t][i][h*32+d] * A[r][h][d][c]   (m = 2*r + isV)
// row==256 encodes the bias fold.
__global__ void fold_kernel(const float* __restrict__ Wk, const float* __restrict__ bk,
                            const float* __restrict__ Wv, const float* __restrict__ bv,
                            const float* __restrict__ rel_att, const float* __restrict__ rel_msg,
                            float* __restrict__ Wf, float* __restrict__ bf) {
    int row = blockIdx.x % 257;
    int m   = blockIdx.x / 257;          // 0..5
    int r   = m >> 1;
    int isV = m & 1;
    int t   = (r == 1) ? 1 : 0;          // src node type per relation {A,B,A}
    int c   = threadIdx.x;               // 0..255 output column
    int h   = c >> 5;
    int cc  = c & 31;
    const float* W = isV ? Wv : Wk;
    const float* B = isV ? bv : bk;
    const float* A = (isV ? rel_msg : rel_att) + ((size_t)r * HH + h) * DKK * DKK;
    float acc = 0.f;
    if (row < 256) {
        const float* wrow = W + ((size_t)t * INF_ + row) * OUTF + h * DKK;
        #pragma unroll
        for (int d = 0; d < DKK; ++d) acc += wrow[d] * A[d * DKK + cc];
        Wf[((size_t)m * INF_ + row) * OUTF + c] = acc;
    } else {
        const float* brow = B + (size_t)t * OUTF + h * DKK;
        #pragma unroll
        for (int d = 0; d < DKK; ++d) acc += brow[d] * A[d * DKK + cc];
        bf[(size_t)m * OUTF + c] = acc;
    }
}

// ------------------------------------------------------------ WMMA f32 GEMM
// Y[16 rows x 256 cols per block] = x_scale*(X @ W) + bias, optional skip-mix
// epilogue: Y = (..)*sigmoid(skip) + h_skip*(1-sigmoid(skip)).
// 8 waves/block; wave w owns cols [32w, 32w+32) as two 16x16 tiles; K=256 in
// steps of 4 via V_WMMA_F32_16X16X4_F32 with C=D accumulation chaining.
// X tile is staged into LDS via the CDNA5 async DMA path (ASYNCcnt).
__global__ __launch_bounds__(256)
void gemm_wmma_kernel(const float* __restrict__ X, const float* __restrict__ W,
                      const float* __restrict__ bias, float* __restrict__ Y,
                      float x_scale,
                      const float* __restrict__ h_skip,
                      const float* __restrict__ skip_scalar) {
    __shared__ float Xs[16 * LDS_STRIDE];
    const int tid = threadIdx.x;
    const int rowbase = blockIdx.x * 16;

    // async stage of the 16x256 X tile (GLOBAL_LOAD_ASYNC_TO_LDS_B128, GV mode:
    // vdst = per-lane LDS byte address (= addr[31:0] of the generic pointer),
    // vaddr = 64-bit global address). 16B per lane per op, all lanes active.
    {
        const char* gbase = (const char*)(X + (size_t)rowbase * OUTF);
        #pragma unroll
        for (int i = tid; i < 1024; i += 256) {
            int rr = i >> 6, cv = (i & 63) << 2;
            unsigned ldsAddr =
                (unsigned)(uintptr_t)(const void*)(Xs + rr * LDS_STRIDE + cv);
            unsigned long long ga =
                (unsigned long long)(uintptr_t)gbase + (unsigned long long)i * 16ull;
            asm volatile("global_load_async_to_lds_b128 %0, %1, off"
                         :: "v"(ldsAddr), "v"(ga)
                         : "memory");
        }
        asm volatile("s_wait_asynccnt 0x0" ::: "memory");
    }
    __syncthreads();

    const int wave  = tid >> 5;
    const int lane  = tid & 31;
    const int lhalf = lane >> 4;     // 0: K+{0,1}, 1: K+{2,3} (ISA f32 A 16x4 layout)
    const int l16   = lane & 15;
    const int c0    = wave * 32;

    v8f acc0 = {};
    v8f acc1 = {};
    const float* xrow = Xs + l16 * LDS_STRIDE + lhalf * 2;

    #pragma unroll 4
    for (int k = 0; k < 256; k += 4) {
        v2f a;
        a.x = xrow[k];
        a.y = xrow[k + 1];
        const float* wbase = W + ((size_t)(k + lhalf * 2)) * OUTF;
        v2f b0, b1;
        b0.x = wbase[c0 + l16];
        b0.y = wbase[OUTF + c0 + l16];
        b1.x = wbase[c0 + 16 + l16];
        b1.y = wbase[OUTF + c0 + 16 + l16];
        acc0 = __builtin_amdgcn_wmma_f32_16x16x4_f32(false, a, false, b0,
                                                     (short)0, acc0, false, false);
        acc1 = __builtin_amdgcn_wmma_f32_16x16x4_f32(false, a, false, b1,
                                                     (short)0, acc1, false, false);
    }

    float alpha = 1.f, beta = 0.f;
    if (h_skip) {
        float s = skip_scalar[0];
        alpha = 1.f / (1.f + __expf(-s));
        beta  = 1.f - alpha;
    }

    #pragma unroll
    for (int rr = 0; rr < 8; ++rr) {
        int row  = rowbase + rr + lhalf * 8;   // ISA f32 16x16 C/D layout
        int col  = c0 + l16;
        float v0 = acc0[rr] * x_scale + bias[col];
        float v1 = acc1[rr] * x_scale + bias[col + 16];
        size_t o = (size_t)row * OUTF + col;
        if (h_skip) {
            Y[o]      = v0 * alpha + h_skip[o]      * beta;
            Y[o + 16] = v1 * alpha + h_skip[o + 16] * beta;
        } else {
            Y[o]      = v0;
            Y[o + 16] = v1;
        }
    }
}

// ------------------------------------------------------------- edge kernels
// one wave per (edge, head); lane = DK element -> 128B coalesced gathers
__global__ __launch_bounds__(256)
void score_kernel(const float* __restrict__ q, const float* __restrict__ k,
                  const int* __restrict__ src, const int* __restrict__ dst,
                  const float* __restrict__ pri, float inv_sqrt_dk,
                  float* __restrict__ score, float* __restrict__ mx, int total) {
    int idx = blockIdx.x * 8 + (threadIdx.x >> 5);
    if (idx >= total) return;
    int lane = threadIdx.x & 31;
    int e = idx >> 3, h = idx & 7;
    int s = src[e], d = dst[e];
    float p = q[(size_t)d * OUTF + h * DKK + lane] *
              k[(size_t)s * OUTF + h * DKK + lane];
    #pragma unroll
    for (int off = 16; off; off >>= 1) p += __shfl_xor(p, off, 32);
    if (lane == 0) {
        p *= pri[h] * inv_sqrt_dk;
        score[idx] = p;
        atomicMaxF(&mx[(size_t)d * HH + h], p);
    }
}

__global__ void expden_kernel(float* __restrict__ score, const int* __restrict__ dst,
                              const float* __restrict__ mx, float* __restrict__ den,
                              int total) {
    int i = blockIdx.x * blockDim.x + threadIdx.x;
    if (i >= total) return;
    int e = i >> 3, h = i & 7;
    int d = dst[e];
    float ex = __expf(score[i] - mx[(size_t)d * HH + h]);
    score[i] = ex;                       // in-place: score buffer becomes exp
    atomicAdd(&den[(size_t)d * HH + h], ex);
}

__global__ __launch_bounds__(256)
void msg_kernel(const float* __restrict__ v, const float* __restrict__ ex,
                const int* __restrict__ src, const int* __restrict__ dst,
                const float* __restrict__ den, float* __restrict__ t, int total) {
    int idx = blockIdx.x * 8 + (threadIdx.x >> 5);
    if (idx >= total) return;
    int lane = threadIdx.x & 31;
    int e = idx >> 3, h = idx & 7;
    int s = src[e], d = dst[e];
    float attn = ex[idx] / den[(size_t)d * HH + h];
    atomicAdd(&t[(size_t)d * OUTF + h * DKK + lane],
              v[(size_t)s * OUTF + h * DKK + lane] * attn);
}

// ------------------------------------------------------------------- launch
extern "C" void kernel_launch(void* const* d_in, const int* in_sizes, int n_in,
                              void* d_out, int out_size, void* d_ws, size_t ws_size,
                              hipStream_t stream) {
    (void)in_sizes; (void)n_in; (void)out_size; (void)ws_size;

    const float* h_A = (const float*)d_in[0];
    const float* h_B = (const float*)d_in[1];
    const int* srcs[3] = {(const int*)d_in[2], (const int*)d_in[4], (const int*)d_in[6]};
    const int* dsts[3] = {(const int*)d_in[3], (const int*)d_in[5], (const int*)d_in[7]};
    const float* Wk = (const float*)d_in[8];
    const float* bk = (const float*)d_in[9];
    const float* Wq = (const float*)d_in[10];
    const float* bq = (const float*)d_in[11];
    const float* Wv = (const float*)d_in[12];
    const float* bv = (const float*)d_in[13];
    const float* Wa = (const float*)d_in[14];
    const float* ba = (const float*)d_in[15];
    const float* rel_att = (const float*)d_in[16];
    const float* rel_msg = (const float*)d_in[17];
    const float* rel_pri = (const float*)d_in[18];
    const float* skip    = (const float*)d_in[19];
    float* out = (float*)d_out;

    // workspace carve-up (floats)
    float* ws = (float*)d_ws;
    const size_t NF = (size_t)NN * OUTF;          // 12.8M floats
    float* qA    = ws;
    float* qB    = qA + NF;
    float* kbuf  = qB + NF;
    float* vbuf  = kbuf + NF;
    float* tA    = vbuf + NF;
    float* tB    = tA + NF;                       // tA,tB contiguous
    float* score = tB + NF;                       // E*H
    float* mx    = score + (size_t)EE * HH;       // N*H
    float* den   = mx + (size_t)NN * HH;
    float* Wf    = den + (size_t)NN * HH;         // 6*256*256
    float* bf    = Wf + (size_t)6 * INF_ * OUTF;  // 6*256

    const int gemm_blocks = NN / 16;              // 3125
    const int totalEH = EE * HH;                  // 6.4M (edge,head) pairs
    const float inv_sqrt_dk = 0.17677669529663687f; // 1/sqrt(32)

    // zero the message accumulators (tA,tB) every call
    fill_kernel<<<dim3((unsigned)((2 * NF + 255) / 256)), dim3(256), 0, stream>>>(
        tA, 0.f, (int)(2 * NF));

    // fold per-relation 32x32 head transforms into K/V projection weights
    fold_kernel<<<dim3(6 * 257), dim3(256), 0, stream>>>(Wk, bk, Wv, bv,
                                                         rel_att, rel_msg, Wf, bf);

    // q projections (per dst type)
    gemm_wmma_kernel<<<dim3(gemm_blocks), dim3(256), 0, stream>>>(
        h_A, Wq, bq, qA, 1.f, nullptr, nullptr);
    gemm_wmma_kernel<<<dim3(gemm_blocks), dim3(256), 0, stream>>>(
        h_B, Wq + (size_t)INF_ * OUTF, bq + OUTF, qB, 1.f, nullptr, nullptr);

    const int srcType[3] = {0, 1, 0};
    float* tTgt[3]       = {tB, tA, tB};
    const float* qTgt[3] = {qB, qA, qB};

    for (int r = 0; r < 3; ++r) {
        const float* hsrc = srcType[r] ? h_B : h_A;
        // k_r, v_r with relation transform folded into the weights
        gemm_wmma_kernel<<<dim3(gemm_blocks), dim3(256), 0, stream>>>(
            hsrc, Wf + (size_t)(2 * r) * INF_ * OUTF, bf + (2 * r) * OUTF,
            kbuf, 1.f, nullptr, nullptr);
        gemm_wmma_kernel<<<dim3(gemm_blocks), dim3(256), 0, stream>>>(
            hsrc, Wf + (size_t)(2 * r + 1) * INF_ * OUTF, bf + (2 * r + 1) * OUTF,
            vbuf, 1.f, nullptr, nullptr);

        // segment softmax state
        fill_kernel<<<dim3((NN * HH + 255) / 256), dim3(256), 0, stream>>>(
            mx, -3.0e38f, NN * HH);
        fill_kernel<<<dim3((NN * HH + 255) / 256), dim3(256), 0, stream>>>(
            den, 0.f, NN * HH);

        score_kernel<<<dim3(totalEH / 8), dim3(256), 0, stream>>>(
            qTgt[r], kbuf, srcs[r], dsts[r], rel_pri + r * HH, inv_sqrt_dk,
            score, mx, totalEH);
        expden_kernel<<<dim3((totalEH + 255) / 256), dim3(256), 0, stream>>>(
            score, dsts[r], mx, den, totalEH);
        msg_kernel<<<dim3(totalEH / 8), dim3(256), 0, stream>>>(
            vbuf, score, srcs[r], dsts[r], den, tTgt[r], totalEH);
    }

    // output transforms with fused sigmoid-skip epilogue
    // t_A = t1 ; t_B = (t0+t2)*0.5 (x_scale folds the mean into the epilogue)
    gemm_wmma_kernel<<<dim3(gemm_blocks), dim3(256), 0, stream>>>(
        tA, Wa, ba, out, 1.f, h_A, skip);
    gemm_wmma_kernel<<<dim3(gemm_blocks), dim3(256), 0, stream>>>(
        tB, Wa + (size_t)OUTF * OUTF, ba + OUTF, out + NF, 0.5f, h_B, skip + 1);
}